// FP64ToFP32Converter_76312978916102
// MI455X (gfx1250) — compile-verified
//
#include <hip/hip_runtime.h>
#include <stdint.h>

// FP64-bit-vector -> FP32-bit-vector converter for MI455X (gfx1250).
// Memory-bound streaming kernel: 384 B per element, ~17.3 us floor @ 23.3 TB/s.
// Data movement uses CDNA5 async global<->LDS instructions for full coalescing.

#define ELEMS_PER_BLOCK 128          // 128 threads = 4 wave32, 1 element/thread
#define ROW_DW   68                  // 64 payload dwords + 4 pad dwords
#define ROW_BYTES (ROW_DW * 4)       // 272 B: 16B-aligned rows, reduced bank conflicts

typedef __attribute__((address_space(3))) uint8_t lds_u8;

// Exact integer re-expression of the reference gate network.
__device__ __forceinline__ uint32_t convert_bits(uint64_t u) {
    uint32_t sign = (uint32_t)(u >> 63) << 31;
    uint32_t e    = (uint32_t)(u >> 52) & 0x7FFu;
    uint64_t m    = u & 0x000FFFFFFFFFFFFFULL;
    if (e == 0x7FFu)                              // NaN / Inf (NaN wins, as in ref mux order)
        return sign | (m ? 0x7FC00000u : 0x7F800000u);
    if (e > 1150u) return sign | 0x7F800000u;     // overflow -> Inf
    if (e < 897u)  return sign;                   // underflow -> signed zero (FTZ)
    uint32_t m23 = (uint32_t)(m >> 29);           // kept mantissa bits (MSB-aligned)
    uint32_t R   = (uint32_t)(m >> 28) & 1u;      // round bit
    uint32_t S   = ((m & 0x0FFFFFFFULL) != 0ull); // sticky (28 bits)
    uint32_t rup = R & (S | (m23 & 1u));          // RNE
    // add into combined exp|mant so a mantissa carry bumps the exponent
    uint32_t me  = (((e - 896u) << 23) | m23) + rup;  // may reach 0x7F800000 == Inf, matches ref
    return sign | me;
}

__global__ __launch_bounds__(ELEMS_PER_BLOCK)
void fp64pulse_to_fp32pulse(const float* __restrict__ in, float* __restrict__ out, int nelem) {
    __shared__ float smem[ELEMS_PER_BLOCK * ROW_DW];   // 34,816 B

    const int tid  = threadIdx.x;
    const int lane = tid & 31;
    const int wave = tid >> 5;
    const long long blockBase = (long long)blockIdx.x * ELEMS_PER_BLOCK;

    if (blockBase + ELEMS_PER_BLOCK <= (long long)nelem) {
        // ---------- Phase 1: async global -> LDS stage (8 KB per wave, coalesced) ----------
        // Raw LDS byte offset of smem (HW adds LDS_BASE for async-to-LDS addressing).
        const uint32_t lds_base = (uint32_t)(uintptr_t)(lds_u8*)(void*)smem;
        const uint32_t wave_lds = lds_base + (uint32_t)(wave * 32 * ROW_BYTES);
        const uint64_t gsrc = (uint64_t)(uintptr_t)(in + (blockBase + (long long)wave * 32) * 64);

#pragma unroll
        for (int j = 0; j < 16; ++j) {
            int c = j * 32 + lane;                 // 16B chunk id in wave tile, 0..511
            int e = c >> 4;                        // element within wave
            int o = (c & 15) * 16;                 // byte offset within element row
            uint64_t gaddr = gsrc + (uint64_t)c * 16;          // contiguous 512B/instr
            uint32_t laddr = wave_lds + (uint32_t)(e * ROW_BYTES + o);
            asm volatile("global_load_async_to_lds_b128 %0, %1, off"
                         :: "v"(laddr), "v"(gaddr) : "memory");
        }
        asm volatile("s_wait_asynccnt 0" ::: "memory");        // loads landed in LDS

        // ---------- Phase 2: pack 64 bits, integer convert ----------
        const float4* row = (const float4*)(smem + (size_t)tid * ROW_DW);
        uint64_t bits = 0;
#pragma unroll
        for (int t = 0; t < 16; ++t) {
            float4 v = row[t];
            // inputs are exactly 0.0f / 1.0f: bit 29 of the pattern distinguishes them
            uint32_t nib = (((__float_as_uint(v.x) >> 29) & 1u) << 3) |
                           (((__float_as_uint(v.y) >> 29) & 1u) << 2) |
                           (((__float_as_uint(v.z) >> 29) & 1u) << 1) |
                           ( (__float_as_uint(v.w) >> 29) & 1u);
            bits = (bits << 4) | nib;              // MSB-first -> standard f64 bit layout
        }
        const uint32_t r = convert_bits(bits);

        // ---------- Phase 3: unpack 32 bits into own LDS row, async LDS -> global ----------
        float4* os = (float4*)(smem + (size_t)tid * ROW_DW);
#pragma unroll
        for (int t = 0; t < 8; ++t) {
            float4 v;
            v.x = (float)((r >> (31 - (t * 4 + 0))) & 1u);
            v.y = (float)((r >> (31 - (t * 4 + 1))) & 1u);
            v.z = (float)((r >> (31 - (t * 4 + 2))) & 1u);
            v.w = (float)((r >> (31 - (t * 4 + 3))) & 1u);
            os[t] = v;
        }
        asm volatile("s_wait_dscnt 0" ::: "memory");           // LDS writes visible to DMA

        const uint64_t gdst = (uint64_t)(uintptr_t)(out + (blockBase + (long long)wave * 32) * 32);
#pragma unroll
        for (int j = 0; j < 8; ++j) {
            int c  = j * 32 + lane;                // 16B chunk id, 0..255 (4 KB per wave)
            int e2 = c >> 3;                       // producer lane / element within wave
            int o2 = (c & 7) * 16;
            uint32_t laddr = wave_lds + (uint32_t)(e2 * ROW_BYTES + o2);
            uint64_t gaddr = gdst + (uint64_t)c * 16;          // contiguous 512B/instr
            asm volatile("global_store_async_from_lds_b128 %0, %1, off"
                         :: "v"(gaddr), "v"(laddr) : "memory");
        }
        asm volatile("s_wait_asynccnt 0" ::: "memory");        // drain before wave exit
    } else {
        // Tail fallback (unused for B = 1M, kept for safety): direct loads/stores.
        long long idx = blockBase + tid;
        if (idx < (long long)nelem) {
            const float* src = in + idx * 64;
            uint64_t bits = 0;
            for (int i = 0; i < 64; ++i)
                bits = (bits << 1) | ((__float_as_uint(src[i]) >> 29) & 1u);
            uint32_t r = convert_bits(bits);
            float* dst = out + idx * 32;
            for (int k = 0; k < 32; ++k)
                dst[k] = (float)((r >> (31 - k)) & 1u);
        }
    }
}

extern "C" void kernel_launch(void* const* d_in, const int* in_sizes, int n_in,
                              void* d_out, int out_size, void* d_ws, size_t ws_size,
                              hipStream_t stream) {
    const float* in = (const float*)d_in[0];
    float* out = (float*)d_out;
    int nelem = in_sizes[0] / 64;                  // B elements of 64 bits each
    int blocks = (nelem + ELEMS_PER_BLOCK - 1) / ELEMS_PER_BLOCK;
    fp64pulse_to_fp32pulse<<<blocks, ELEMS_PER_BLOCK, 0, stream>>>(in, out, nelem);
}